// DifferentiableTopK_22746146799827
// MI455X (gfx1250) — compile-verified
//
#include <hip/hip_runtime.h>
#include <stdint.h>

// Problem constants (from reference): B=64, D=131072, K=5.
#define NB 64
#define ND 131072
#define NK 5

#define TPB 512                 // 16 wave32 waves per block (one block per row)
#define NWAVES (TPB / 32)       // 16
#define SEG (ND / NWAVES)       // 8192 floats owned by each wave (contiguous)
#define CHUNK 128               // floats per async chunk per wave (32 lanes x b128)
#define NCHUNK (SEG / CHUNK)    // 64 chunks per wave
#define NSLOT 4                 // LDS ring depth per wave

typedef __attribute__((ext_vector_type(4))) float v4f;
typedef __attribute__((ext_vector_type(4))) int v4i;

// ---- CDNA5 async global->LDS path (guarded; fallback = NT vector loads) ----
#if defined(__has_builtin)
#if __has_builtin(__builtin_amdgcn_global_load_async_to_lds_b128)
#define HAVE_ASYNC_LDS 1
#endif
#if __has_builtin(__builtin_amdgcn_s_wait_asynccnt)
#define WAIT_ASYNC(n) __builtin_amdgcn_s_wait_asynccnt(n)
#endif
#endif
#ifndef WAIT_ASYNC
#define WAIT_ASYNC(n) asm volatile("s_wait_asynccnt %0" ::"n"(n) : "memory")
#endif

// Builtin signature (from diagnostics): (v4i addrspace(1)*, v4i addrspace(3)*, imm, imm)
__device__ static inline __attribute__((address_space(3))) v4i* as_lds(void* p) {
  return (__attribute__((address_space(3))) v4i*)(uint32_t)(uintptr_t)p;
}
__device__ static inline __attribute__((address_space(1))) v4i* as_glob(const void* p) {
  return (__attribute__((address_space(1))) v4i*)(uintptr_t)p;
}

// Monotonic 64-bit key: (orderable float bits << 32) | ~index
// -> bigger key = bigger value, ties broken toward LOWER index (jax top_k semantics).
__device__ static inline unsigned long long packKey(float v, uint32_t idx) {
  uint32_t u = __float_as_uint(v);
  u = (u & 0x80000000u) ? ~u : (u | 0x80000000u);
  return ((unsigned long long)u << 32) | (unsigned long long)(0xFFFFFFFFu - idx);
}

__device__ static inline unsigned long long waveMax64(unsigned long long m) {
#pragma unroll
  for (int off = 16; off > 0; off >>= 1) {
    unsigned long long o = __shfl_xor(m, off, 32);
    if (o > m) m = o;
  }
  return m;
}

// Each lane holds a sorted (descending) 5-key list; 5 rounds of
// "wave-argmax + pop winner" yield the merged wave top-5 (known to all lanes).
__device__ static inline void merge5(unsigned long long c0, unsigned long long c1,
                                     unsigned long long c2, unsigned long long c3,
                                     unsigned long long c4,
                                     unsigned long long res[NK]) {
  int p = 0;
#pragma unroll
  for (int k = 0; k < NK; ++k) {
    unsigned long long cand =
        (p == 0) ? c0 : (p == 1) ? c1 : (p == 2) ? c2 : (p == 3) ? c3 : (p == 4) ? c4 : 0ull;
    unsigned long long m = waveMax64(cand);
    res[k] = m;
    if (cand == m && m != 0ull) p++;
  }
}

#define SWAP_PAIR(a, b, x, y) \
  { float _t = (a); (a) = (b); (b) = _t; uint32_t _u = (x); (x) = (y); (y) = _u; }

// Insert (val, idx) into the sorted-descending 5-element register list.
// Strict '>' keeps earlier (lower) indices ahead on exact ties.
#define TRY_INSERT(val, idx)                              \
  do {                                                    \
    float _v = (val); uint32_t _i = (idx);                \
    if (_v > tv4) {                                       \
      tv4 = _v; ti4 = _i;                                 \
      if (tv4 > tv3) SWAP_PAIR(tv3, tv4, ti3, ti4);       \
      if (tv3 > tv2) SWAP_PAIR(tv2, tv3, ti2, ti3);       \
      if (tv2 > tv1) SWAP_PAIR(tv1, tv2, ti1, ti2);       \
      if (tv1 > tv0) SWAP_PAIR(tv0, tv1, ti0, ti1);       \
    }                                                     \
  } while (0)

// ---------------- Kernel 1: zero-fill d_out with NT b128 stores ----------------
__global__ __launch_bounds__(256) void dtopk_zero_fill(v4f* __restrict__ p, long n4) {
  long i = (long)blockIdx.x * blockDim.x + threadIdx.x;
  long stride = (long)gridDim.x * blockDim.x;
  v4f z = (v4f)0.0f;
  for (; i < n4; i += stride) __builtin_nontemporal_store(z, p + i);
}

// ---------------- Kernel 2: per-row top-5 + one-hot scatter ----------------
__global__ __launch_bounds__(TPB) void dtopk_topk_onehot(const float* __restrict__ sims,
                                                         float* __restrict__ out) {
  __shared__ float ldsBuf[NWAVES * NSLOT * CHUNK];      // 32 KB stream ring
  __shared__ unsigned long long wkeys[NWAVES * NK];     // per-wave merged top-5

  const int row = blockIdx.x;
  const int tid = (int)threadIdx.x;
  const int lane = tid & 31;
  const int wave = tid >> 5;

  const float* g = sims + (size_t)row * ND + (size_t)wave * SEG;
  float* myLds = ldsBuf + wave * (NSLOT * CHUNK);
  const uint32_t colBase = (uint32_t)(wave * SEG) + (uint32_t)(lane * 4);

  float tv0 = -__builtin_inff(), tv1 = -__builtin_inff(), tv2 = -__builtin_inff(),
        tv3 = -__builtin_inff(), tv4 = -__builtin_inff();
  uint32_t ti0 = 0, ti1 = 0, ti2 = 0, ti3 = 0, ti4 = 0;

#define PROC_CHUNK(c)                                                     \
  do {                                                                    \
    int _s = (c) & (NSLOT - 1);                                           \
    v4f _d = *(const v4f*)(myLds + _s * CHUNK + lane * 4);                \
    uint32_t _col = colBase + (uint32_t)(c) * CHUNK;                      \
    TRY_INSERT(_d.x, _col);                                               \
    TRY_INSERT(_d.y, _col + 1);                                           \
    TRY_INSERT(_d.z, _col + 2);                                           \
    TRY_INSERT(_d.w, _col + 3);                                           \
  } while (0)

#ifdef HAVE_ASYNC_LDS
  // Prologue: fill the 4-slot ring (per-wave private => no block barriers).
#pragma unroll
  for (int s = 0; s < NSLOT; ++s)
    __builtin_amdgcn_global_load_async_to_lds_b128(
        as_glob(g + s * CHUNK + lane * 4),
        as_lds(myLds + s * CHUNK + lane * 4), 0, 0);

  // Steady state: wait oldest done, consume it, refill that slot.
  for (int c = 0; c < NCHUNK - NSLOT; ++c) {
    WAIT_ASYNC(NSLOT - 1);
    PROC_CHUNK(c);
    int slot = c & (NSLOT - 1);
    __builtin_amdgcn_global_load_async_to_lds_b128(
        as_glob(g + (c + NSLOT) * CHUNK + lane * 4),
        as_lds(myLds + slot * CHUNK + lane * 4), 0, 0);
  }
  // Drain tail (immediates required by s_wait_asynccnt).
  WAIT_ASYNC(3); PROC_CHUNK(NCHUNK - 4);
  WAIT_ASYNC(2); PROC_CHUNK(NCHUNK - 3);
  WAIT_ASYNC(1); PROC_CHUNK(NCHUNK - 2);
  WAIT_ASYNC(0); PROC_CHUNK(NCHUNK - 1);
#else
  // Fallback: direct non-temporal b128 streaming loads.
  for (int c = 0; c < NCHUNK; ++c) {
    v4f d = __builtin_nontemporal_load((const v4f*)(g + c * CHUNK + lane * 4));
    uint32_t col = colBase + (uint32_t)c * CHUNK;
    TRY_INSERT(d.x, col);
    TRY_INSERT(d.y, col + 1);
    TRY_INSERT(d.z, col + 2);
    TRY_INSERT(d.w, col + 3);
  }
#endif

  // Wave-level merge of 32 lane-local top-5 lists.
  unsigned long long fin[NK];
  merge5(packKey(tv0, ti0), packKey(tv1, ti1), packKey(tv2, ti2),
         packKey(tv3, ti3), packKey(tv4, ti4), fin);
  if (lane == 0) {
#pragma unroll
    for (int k = 0; k < NK; ++k) wkeys[wave * NK + k] = fin[k];
  }
  __syncthreads();

  // Block-level merge: wave 0 merges the 16 per-wave lists.
  if (wave == 0) {
    unsigned long long m0 = 0, m1 = 0, m2 = 0, m3 = 0, m4 = 0;
    if (lane < NWAVES) {
      m0 = wkeys[lane * NK + 0];
      m1 = wkeys[lane * NK + 1];
      m2 = wkeys[lane * NK + 2];
      m3 = wkeys[lane * NK + 3];
      m4 = wkeys[lane * NK + 4];
    }
    unsigned long long res[NK];
    merge5(m0, m1, m2, m3, m4, res);
    if (lane == 0) {
#pragma unroll
      for (int k = 0; k < NK; ++k) {
        uint32_t idx = 0xFFFFFFFFu - (uint32_t)res[k];
        // Output layout: [B*K] hard indices, then [B,K,D] one-hot weights.
        out[row * NK + k] = (float)idx;
        size_t off = (size_t)(NB * NK) + (size_t)(row * NK + k) * (size_t)ND + idx;
        out[off] = 1.0f;
      }
    }
  }
#undef PROC_CHUNK
}

extern "C" void kernel_launch(void* const* d_in, const int* in_sizes, int n_in,
                              void* d_out, int out_size, void* d_ws, size_t ws_size,
                              hipStream_t stream) {
  (void)in_sizes; (void)n_in; (void)out_size; (void)d_ws; (void)ws_size;
  const float* sims = (const float*)d_in[0];
  float* out = (float*)d_out;

  // Total output floats = B*K (indices) + B*K*D (one-hot) = 41,943,360 (divisible by 4).
  const long n4 = ((long)NB * NK * ND + (long)NB * NK) / 4;
  dtopk_zero_fill<<<4096, 256, 0, stream>>>((v4f*)out, n4);
  dtopk_topk_onehot<<<NB, TPB, 0, stream>>>(sims, out);
}